// SchoolImage_52218212384769
// MI455X (gfx1250) — compile-verified
//
#include <hip/hip_runtime.h>

#define N_PTS 8192
#define DDIM 64
#define KNN 32
#define KP1 64
#define PROJ_ITERS 100

typedef __attribute__((ext_vector_type(2))) float v2f;
typedef __attribute__((ext_vector_type(4))) float v4f;
typedef __attribute__((ext_vector_type(8))) float v8f;

__device__ __forceinline__ float wave_sum(float v) {
    #pragma unroll
    for (int off = 16; off > 0; off >>= 1)
        v += __shfl_xor(v, off, 32);
    return v;
}

__global__ __launch_bounds__(256) void SchoolImage_fused_kernel(
    const float* __restrict__ Y,
    const float* __restrict__ S,      // semantic_H
    const float* __restrict__ O,      // ortho_H
    const int*   __restrict__ idx,
    const int*   __restrict__ alphaP,
    const int*   __restrict__ betaP,
    float* __restrict__ out_embs,
    float* __restrict__ out_A,
    float* __restrict__ out_Y)
{
    __shared__ float smem[8][KNN * DDIM];   // 8 waves * 8KB = 64KB
    const int lane = threadIdx.x & 31;
    const int wv   = threadIdx.x >> 5;
    const int row  = blockIdx.x * 8 + wv;   // one wave32 per row

    const float alpha = (float)alphaP[0];
    const float beta  = (float)betaP[0];

    // ---- Zero this wave's A row with wide NON-TEMPORAL streaming stores.
    // A is 268 MB write-once: NT keeps it from rinsing the 192 MB L2 where
    // the 6 MB of gather tables (Y/O/S, each reused ~32x) should stay.
    // Stores retire in the background while the wave does distances +
    // projection; same-wave store->store order per address is maintained, so
    // the later scatter of w into this row is ordered correctly.
    {
        v4f z = {0.f, 0.f, 0.f, 0.f};
        v4f* arow = (v4f*)(out_A + (size_t)row * N_PTS);
        #pragma unroll
        for (int q = 0; q < (N_PTS / 4) / 32; ++q)   // 64 float4 per lane
            __builtin_nontemporal_store(z, &arow[q * 32 + lane]);
    }

    // ---- Pass-through copy of Y row (third output).
    {
        const v4f* yi4 = (const v4f*)(Y + (size_t)row * DDIM);
        v4f* oy4 = (v4f*)(out_Y + (size_t)row * DDIM);
        if (lane < 16) __builtin_nontemporal_store(yi4[lane], &oy4[lane]);
    }

    // ---- idxa0 = idx[:, 1:33]; one neighbor per lane.
    const int nbr = idx[(size_t)row * KP1 + 1 + lane];

    // ---- Distances dfi (Y) and dxi (ortho_H), D = 64 via 16x float4.
    float accF = 0.f, accX = 0.f;
    {
        const float4* yi = (const float4*)(Y + (size_t)row * DDIM);
        const float4* yn = (const float4*)(Y + (size_t)nbr * DDIM);
        const float4* oi = (const float4*)(O + (size_t)row * DDIM);
        const float4* on = (const float4*)(O + (size_t)nbr * DDIM);
        #pragma unroll
        for (int q = 0; q < 16; ++q) {
            float4 a = yi[q], b = yn[q];
            float dx = a.x - b.x, dy = a.y - b.y, dz = a.z - b.z, dw = a.w - b.w;
            accF += dx*dx + dy*dy + dz*dz + dw*dw;
            float4 c = oi[q], d = on[q];
            dx = c.x - d.x; dy = c.y - d.y; dz = c.z - d.z; dw = c.w - d.w;
            accX += dx*dx + dy*dy + dz*dz + dw*dw;
        }
    }
    const float dfi = sqrtf(accF + 1e-8f);
    const float dxi = sqrtf(accX + 1e-8f);
    const float ad  = -(dxi + beta * dfi) / (2.f * alpha);

    // ---- Simplex projection (k=1, n=32), 100 fixed Newton iterations,
    // done wave-resident with ballot + shuffle-tree reductions.
    const float mean = wave_sum(ad) * (1.f / 32.f);
    const float v0   = ad - mean + (1.f / 32.f);
    float lam = 0.f;
    for (int it = 0; it < PROJ_ITERS; ++it) {
        float v1 = v0 - lam;
        bool  pos = v1 > 0.f;
        int   npos = __popcll(__ballot(pos));
        if (npos < 1) npos = 1;
        float f = wave_sum(pos ? v1 : 0.f) - 1.f;
        lam += f / (float)npos;
    }
    const float w = fmaxf(v0 - lam, 0.f);

    // ---- Duplicate-column dedupe: .at[].set => last write wins.
    bool dup = false;
    #pragma unroll
    for (int t = 0; t < 32; ++t) {
        int o = __shfl(nbr, t, 32);
        if (t > lane && o == nbr) dup = true;
    }
    const float weff = dup ? 0.f : w;

    // ---- Scatter w into A (row already zeroed by this wave; NT as well).
    if (!dup) __builtin_nontemporal_store(w, &out_A[(size_t)row * N_PTS + nbr]);

    // ---- Stage gathered semantic_H rows (32 x 64 f32) into LDS.
    {
        const float4* sn = (const float4*)(S + (size_t)nbr * DDIM);
        float4* dst = (float4*)(&smem[wv][lane * DDIM]);
        #pragma unroll
        for (int q = 0; q < 16; ++q) dst[q] = sn[q];
    }
    __syncthreads();

    // ---- embs_hom[row] = sum_j weff[j] * S[nbr_j]  via V_WMMA_F32_16X16X4_F32.
    // A-frag (16x4 f32): VGPR0 holds K={0 | 2}, VGPR1 holds K={1 | 3} split by
    // lane half; we replicate w over M. B-frag (4x16 f32): VGPR g holds row
    // K=g (lanes 0-15) / K=2+g (lanes 16-31), N = lane&15. D row 0 carries the
    // 16 output dims of this chunk (all M rows identical by construction).
    const float* srow = smem[wv];
    const int klo = (lane < 16) ? 0 : 2;
    const int col = lane & 15;
    for (int chunk = 0; chunk < 4; ++chunk) {
        v8f c = {0.f, 0.f, 0.f, 0.f, 0.f, 0.f, 0.f, 0.f};
        #pragma unroll
        for (int t = 0; t < 8; ++t) {       // K = 32 neighbors, 4 per step
            v2f a, b;
            a.x = __shfl(weff, 4 * t + klo,     32);
            a.y = __shfl(weff, 4 * t + klo + 1, 32);
            b.x = srow[(4 * t + klo)     * DDIM + chunk * 16 + col];
            b.y = srow[(4 * t + klo + 1) * DDIM + chunk * 16 + col];
            c = __builtin_amdgcn_wmma_f32_16x16x4_f32(
                    /*neg_a=*/false, a, /*neg_b=*/false, b,
                    /*c_mod=*/(short)0, c, /*reuse_a=*/false, /*reuse_b=*/false);
        }
        if (lane < 16)
            out_embs[(size_t)row * DDIM + chunk * 16 + lane] = c[0];
    }
}

extern "C" void kernel_launch(void* const* d_in, const int* in_sizes, int n_in,
                              void* d_out, int out_size, void* d_ws, size_t ws_size,
                              hipStream_t stream) {
    (void)in_sizes; (void)n_in; (void)out_size; (void)d_ws; (void)ws_size;
    const float* Y   = (const float*)d_in[0];
    const float* S   = (const float*)d_in[1];
    const float* O   = (const float*)d_in[2];
    const int*   idx = (const int*)d_in[3];
    const int*   alpha = (const int*)d_in[4];
    const int*   beta  = (const int*)d_in[5];

    float* out      = (float*)d_out;
    float* out_embs = out;                                  // 8192*64
    float* out_A    = out + (size_t)N_PTS * DDIM;           // 8192*8192
    float* out_Y    = out_A + (size_t)N_PTS * N_PTS;        // 8192*64

    SchoolImage_fused_kernel<<<N_PTS / 8, 256, 0, stream>>>(
        Y, S, O, idx, alpha, beta, out_embs, out_A, out_Y);
}